// SoftGCD_29068338659573
// MI455X (gfx1250) — compile-verified
//
#include <hip/hip_runtime.h>
#include <math.h>

// SoftGCD for MI455X (gfx1250).
// Latency-bound nano-kernel: 264 B in, 4 B out. Strategy:
//   - one workgroup of 64 threads = 2 waves on one WGP
//   - wave 0 computes soft_exponent(n1, p_lane), wave 1 computes
//     soft_exponent(n2, p_lane) concurrently  -> halves the serial section
//   - exchange via LDS, one barrier
//   - wave 0 does the 2-element pair soft-sort and reduces the 32-lane dot
//     product  sum(min_exp * log_p)  with one exact-f32
//     V_WMMA_F32_16X16X4_F32 (+ xor-16 shuffle), lane 0 stores exp(total).

typedef __attribute__((ext_vector_type(2))) float v2f;
typedef __attribute__((ext_vector_type(8))) float v8f;

#define SG_ALPHA 200.0f
#define SG_BETA  2560.0f
#define SG_TAU   0.01f
#define SG_MAXK  19

// Soft exponent v_p(n): mirrors _soft_exponents for one prime.
// Soft-sort's last element via the exact min-max isotonic formula at index
// n-1:  out[n-1] = -max_j mean(z[j..n-1]) - w[n-1],
//       z[i] = -desc(x)[i] - w[i], w[i] = (n-i)/tau.
__device__ __forceinline__ float sg_soft_exponent(float n, float p, float logp) {
    float ratio = n / p;
    float frac  = ratio - rintf(ratio);
    float t  = expf(-SG_ALPHA * frac * frac);
    float t2 = t * t, t4 = t2 * t2, t8 = t4 * t4;        // mask = exp(..)**8
    float denz = ratio + ((ratio == 0.0f) ? 1.0f : 0.0f);
    float mask = t8 * (ratio / denz);
    float max_k = floorf(logf(n) / logp);

    float sc[SG_MAXK];
#pragma unroll
    for (int k = 1; k <= SG_MAXK; ++k) {
        float kf = (float)k;
        float r  = n / powf(p, kf);                      // matches primes**ks (f32 pow)
        float fr = r - rintf(r);
        float mk = expf(-SG_BETA * fr * fr);
        sc[k - 1] = (kf <= max_k) ? (mk * kf) : 0.0f;
    }

    // Bitonic sort on 32 (pad with FLT_MAX), ascending; fully unrolled so the
    // array stays in VGPRs and each layer's 16 CEs are independent (short
    // dependency chain, good dual-issue on the SIMD32 VALU).
    float a[32];
#pragma unroll
    for (int i = 0; i < SG_MAXK; ++i) a[i] = sc[i];
#pragma unroll
    for (int i = SG_MAXK; i < 32; ++i) a[i] = 3.402823466e38f;  // +FLT_MAX pads
#pragma unroll
    for (int k = 2; k <= 32; k <<= 1) {
#pragma unroll
        for (int j = k >> 1; j > 0; j >>= 1) {
#pragma unroll
            for (int i = 0; i < 32; ++i) {
                int ij = i ^ j;
                if (ij > i) {
                    bool up = ((i & k) == 0);
                    float x = a[i], y = a[ij];
                    float lo = fminf(x, y), hi = fmaxf(x, y);
                    a[i]  = up ? lo : hi;
                    a[ij] = up ? hi : lo;
                }
            }
        }
    }
    // desc[i] = a[SG_MAXK-1-i]  (pads sit at a[19..31])

    // cumsum of z in sorted-descending order (same op order as jnp.cumsum),
    // then max over suffix means.
    float cs[SG_MAXK + 1];
    cs[0] = 0.0f;
#pragma unroll
    for (int i = 0; i < SG_MAXK; ++i) {
        float w = (float)(SG_MAXK - i) / SG_TAU;
        float z = (-a[SG_MAXK - 1 - i]) - w;
        cs[i + 1] = cs[i] + z;
    }
    float v_last = -3.402823466e38f;
#pragma unroll
    for (int j = 0; j < SG_MAXK; ++j) {
        float mean = (cs[SG_MAXK] - cs[j]) / (float)(SG_MAXK - j);
        v_last = fmaxf(v_last, mean);
    }
    float w_last = 1.0f / SG_TAU;
    float k_hat  = -v_last - w_last;

    return (mask >= 0.99f && max_k >= 1.0f) ? k_hat : 0.0f;
}

__global__ __launch_bounds__(64)
void SoftGCD_29068338659573_kernel(const float* __restrict__ n1p,
                                   const float* __restrict__ n2p,
                                   const float* __restrict__ primes,
                                   const float* __restrict__ logps,
                                   float* __restrict__ out) {
    __shared__ float es[64];

    const int tid  = threadIdx.x;            // 0..63 (2 waves)
    const int lane = tid & 31;               // prime index
    const float p  = primes[lane];
    const float lg = logps[lane];

    // wave 0 handles n1, wave 1 handles n2 (wave-uniform select)
    const float n = (tid < 32) ? n1p[0] : n2p[0];
    es[tid] = sg_soft_exponent(n, p, lg);
    __syncthreads();

    if (tid < 32) {                           // wave-uniform: EXEC all-1s in wave 0
        const float e1 = es[lane];
        const float e2 = es[lane + 32];

        // Pair soft-sort, element 0 (exact min-max formula at index 0):
        // v[0] = min(prefix means of z), out[0] = -v[0] - w[0].
        const float d0 = fmaxf(e1, e2);       // desc[0]
        const float d1 = fminf(e1, e2);       // desc[1]
        const float w0 = 2.0f / SG_TAU;
        const float w1 = 1.0f / SG_TAU;
        const float z0 = -d0 - w0;
        const float z1 = -d1 - w1;
        const float cs1 = z0;                 // cumsum mirror
        const float cs2 = cs1 + z1;
        const float v0  = fminf(cs1, cs2 / 2.0f);
        const float min_exp = -v0 - w0;

        const float s = min_exp * lg;         // per-lane term of the dot product

        // Wave-wide exact f32 sum via one V_WMMA_F32_16X16X4_F32:
        //   A[m][0] = s(lane m), A[m][2] = s(lane m+16), A[*][1]=A[*][3]=0
        //   B = all ones  =>  D[m][n] = s_m + s_{m+16}  for every n.
        // Per-lane sum of the 8 D VGPRs covers one M-half; xor-16 adds the
        // other half => every lane holds the full 32-lane sum.
        v2f a; a[0] = s;    a[1] = 0.0f;
        v2f b; b[0] = 1.0f; b[1] = 1.0f;
        v8f c = {};
        v8f d = __builtin_amdgcn_wmma_f32_16x16x4_f32(
            /*neg_a=*/false, a, /*neg_b=*/false, b,
            /*c_mod=*/(short)0, c, /*reuse_a=*/false, /*reuse_b=*/false);

        float partial = d[0] + d[1] + d[2] + d[3] + d[4] + d[5] + d[6] + d[7];
        float total   = partial + __shfl_xor(partial, 16, 32);

        if (lane == 0) {
            out[0] = expf(total);
        }
    }
}

extern "C" void kernel_launch(void* const* d_in, const int* in_sizes, int n_in,
                              void* d_out, int out_size, void* d_ws, size_t ws_size,
                              hipStream_t stream) {
    (void)in_sizes; (void)n_in; (void)out_size; (void)d_ws; (void)ws_size;
    const float* n1     = (const float*)d_in[0];
    const float* n2     = (const float*)d_in[1];
    const float* primes = (const float*)d_in[2];
    const float* logps  = (const float*)d_in[3];
    float* out = (float*)d_out;

    SoftGCD_29068338659573_kernel<<<dim3(1), dim3(64), 0, stream>>>(
        n1, n2, primes, logps, out);
}